// QuantumGenerator_42125039239637
// MI455X (gfx1250) — compile-verified
//
#include <hip/hip_runtime.h>

typedef float v2f __attribute__((ext_vector_type(2)));
typedef float v8f __attribute__((ext_vector_type(8)));

#define NQ      20
#define OUT_DIM 512
#define FAN_IN  21          // QOUT + COND_DIM
#define BATCH   262144
#define TILES_I 4           // i-tiles (16 rows each) per wave
#define KPAD    24          // K padded for 6x WMMA 16x16x4

// ---------------------------------------------------------------------------
// Kernel 1: exact closed-form <Z_m> for the 2-layer RY + CZ-chain ansatz.
//   <Z_m> = cos(t1_m) cos(t2_m) - sin(t1_m) sin(t2_m) * cos(t1_{m-1}) * cos(t1_{m+1})
// (boundary cos terms = 1; the final CZ layer never affects probabilities)
// Writes a KPAD-entry buffer: qout[0..19], zeros at [20..23].
// ---------------------------------------------------------------------------
__global__ void qsim_expectations(const float* __restrict__ params,
                                  float* __restrict__ qout24) {
    int m = threadIdx.x;
    if (m < KPAD) {
        float v = 0.0f;
        if (m < NQ) {
            float t1 = params[m];         // layer 0, wire m
            float t2 = params[NQ + m];    // layer 1, wire m
            float cL = (m > 0)      ? cosf(params[m - 1]) : 1.0f;
            float cR = (m < NQ - 1) ? cosf(params[m + 1]) : 1.0f;
            v = cosf(t1) * cosf(t2) - sinf(t1) * sinf(t2) * cL * cR;
        }
        qout24[m] = v;
    }
}

// ---------------------------------------------------------------------------
// Branchless tanh: exact identity tanh(x) = 1 - 2/(exp(2x)+1), computed with
// raw v_exp_f32 / v_rcp_f32 (2 trans + 3 VALU). For |x| < 0.25 a 4-term odd
// Taylor polynomial is selected (v_cndmask) to keep relative accuracy near 0.
// Handles large |x| (exp -> inf/0 => result -> +/-1) and NaN correctly.
// ---------------------------------------------------------------------------
__device__ __forceinline__ float fast_tanh(float x) {
    float t    = x * 2.8853900817779268f;          // 2*log2(e)
    float e    = __builtin_amdgcn_exp2f(t);        // v_exp_f32 (full range)
    float re   = __builtin_amdgcn_rcpf(e + 1.0f);  // v_rcp_f32
    float rbig = __builtin_fmaf(-2.0f, re, 1.0f);
    float x2   = x * x;
    float p    = __builtin_fmaf(x2, -0.05396825396825397f, 0.13333333333333333f);
    p          = __builtin_fmaf(x2, p, -0.3333333333333333f);
    float rsm  = __builtin_fmaf(x * x2, p, x);
    return (__builtin_fabsf(x) < 0.25f) ? rsm : rbig;
}

// ---------------------------------------------------------------------------
// Kernel 2: out[i][j] = tanh( [qout(20), cond_i] . W[j][:] + b[j] )
// One wave computes TILES_I vertically adjacent 16x16 output tiles via
// V_WMMA_F32_16X16X4_F32 (K = 21 zero-padded to 24, bias in the accumulator).
// A(kt=0..4) and all B operands are tile-invariant -> loaded once per wave.
//
// Operand layouts (ISA 7.12.2, wave32):
//   A (16x4 f32):  A[m][k] in vgpr (k&1), lane = m + 16*(k>>1)
//   B (4x16 f32):  B[k][n] in vgpr (k&1), lane = n + 16*(k>>1)
//   C/D (16x16):   vgpr r, lane l -> M = r + 8*(l>=16), N = l&15
// ---------------------------------------------------------------------------
__global__ void __launch_bounds__(256)
gen_wmma(const float* __restrict__ cond,
         const float* __restrict__ qout24,
         const float* __restrict__ W,
         const float* __restrict__ b,
         float* __restrict__ out) {
    const int wave = threadIdx.x >> 5;
    const int lane = threadIdx.x & 31;
    const int gw   = blockIdx.x * 8 + wave;
    const int jt   = gw & 31;                 // 32 column tiles (512/16)
    const int ig   = gw >> 5;                 // 4096 row groups (262144/64)
    const int j0   = jt * 16;
    const int i0   = ig * (16 * TILES_I);

    const int half = lane >> 4;               // selects K sub-pair
    const int lm   = lane & 15;               // M for A, N for B
    const int j    = j0 + lm;

    // A operands for kt = 0..4 (k < 20): direct 8B loads from the padded
    // x-vector (broadcast rows are identical) -- plain address math, no
    // register-array indexing for the compiler to scalarize.
    v2f a[5];
#pragma unroll
    for (int kt = 0; kt < 5; ++kt) {
        a[kt] = *(const v2f*)(qout24 + 4 * kt + 2 * half);
    }

    // B operands: column j of W^T (row j of W), zero-padded past k = 20.
    const float* wrow = W + j * FAN_IN;
    v2f bm[6];
#pragma unroll
    for (int kt = 0; kt < 5; ++kt) {
        const int k0 = 4 * kt + 2 * half;
        bm[kt][0] = wrow[k0];
        bm[kt][1] = wrow[k0 + 1];
    }
    const float w20 = wrow[20];               // valid address for every lane
    bm[5][0] = half ? 0.0f : w20;             // cndmask, no divergent branch
    bm[5][1] = 0.0f;

    const float bias = b[j];

#pragma unroll
    for (int t = 0; t < TILES_I; ++t) {
        const int ibase   = i0 + t * 16;
        const float condv = cond[ibase + lm]; // x[i][20], row = i0..+lm
        v2f a5;
        a5[0] = half ? 0.0f : condv;
        a5[1] = 0.0f;

        v8f c;
#pragma unroll
        for (int r = 0; r < 8; ++r) c[r] = bias;

        // 8 args: (neg_a, A, neg_b, B, c_mod, C, reuse_a, reuse_b)
        c = __builtin_amdgcn_wmma_f32_16x16x4_f32(false, a[0], false, bm[0], (short)0, c, false, false);
        c = __builtin_amdgcn_wmma_f32_16x16x4_f32(false, a[1], false, bm[1], (short)0, c, false, false);
        c = __builtin_amdgcn_wmma_f32_16x16x4_f32(false, a[2], false, bm[2], (short)0, c, false, false);
        c = __builtin_amdgcn_wmma_f32_16x16x4_f32(false, a[3], false, bm[3], (short)0, c, false, false);
        c = __builtin_amdgcn_wmma_f32_16x16x4_f32(false, a[4], false, bm[4], (short)0, c, false, false);
        c = __builtin_amdgcn_wmma_f32_16x16x4_f32(false, a5,   false, bm[5], (short)0, c, false, false);

        float* orow = out + (size_t)(ibase + half * 8) * OUT_DIM + j;
#pragma unroll
        for (int r = 0; r < 8; ++r) {
            orow[(size_t)r * OUT_DIM] = fast_tanh(c[r]);
        }
    }
}

extern "C" void kernel_launch(void* const* d_in, const int* in_sizes, int n_in,
                              void* d_out, int out_size, void* d_ws, size_t ws_size,
                              hipStream_t stream) {
    (void)in_sizes; (void)n_in; (void)out_size; (void)ws_size;
    const float* cond   = (const float*)d_in[0];   // (BATCH, 1)
    const float* params = (const float*)d_in[1];   // (2, 20)
    const float* W      = (const float*)d_in[2];   // (512, 21)
    const float* b      = (const float*)d_in[3];   // (512,)
    float*       out    = (float*)d_out;           // (BATCH, 512)
    float*       qout24 = (float*)d_ws;            // KPAD floats scratch

    qsim_expectations<<<1, 32, 0, stream>>>(params, qout24);

    const int waves  = (BATCH / (16 * TILES_I)) * (OUT_DIM / 16); // 131072
    const int blocks = waves / 8;                                 // 16384
    gen_wmma<<<blocks, 256, 0, stream>>>(cond, qout24, W, b, out);
}